// PatchLossBaseline_43052752175785
// MI455X (gfx1250) — compile-verified
//
#include <hip/hip_runtime.h>

// CDNA5 / gfx1250: wave32, fp32 WMMA + Tensor Data Mover async staging.
typedef __attribute__((ext_vector_type(2))) float v2f;
typedef __attribute__((ext_vector_type(8))) float v8f;
typedef __attribute__((ext_vector_type(4))) unsigned int u32x4;
typedef __attribute__((ext_vector_type(4))) int i32x4;
typedef __attribute__((ext_vector_type(8))) int i32x8;

#define B_ 8
#define T_ 1024
#define D_ 384
#define K_ 4

#if defined(__AMDGCN__) && __has_builtin(__builtin_amdgcn_tensor_load_to_lds)
#define USE_TDM 1
#else
#define USE_TDM 0
#endif

__global__ void PatchLoss_zero_kernel(float* __restrict__ out) {
    out[0] = 0.0f;
}

// sq[b*T + t] = sum_d z[b,t,d]^2  -- one wave (32 lanes) per row, 8 rows/block.
__global__ __launch_bounds__(256) void PatchLoss_rowsq_kernel(
    const float* __restrict__ z, float* __restrict__ sq) {
    const int row  = blockIdx.x * 8 + (threadIdx.x >> 5);
    const int lane = threadIdx.x & 31;
    const float* zp = z + (size_t)row * D_;
    float s = 0.0f;
    for (int d = lane; d < D_; d += 32) {
        float v = zp[d];
        s += v * v;
    }
#pragma unroll
    for (int off = 16; off > 0; off >>= 1) s += __shfl_xor(s, off, 32);
    if (lane == 0) sq[row] = s;
}

#if USE_TDM
// TDM: async-DMA an 8-row x 64-float tile of z into LDS with row padding
// 64 DWORDs data + 4 DWORDs pad  ->  68-float LDS row stride (bank stagger).
// D# layout per CDNA5 ISA. Descriptor words are
// wave-uniform; readfirstlane pins the non-constant ones to SGPRs.
__device__ __forceinline__ void tdm_load_rows8(const float* gsrc,
                                               unsigned ldsByteOff) {
    const unsigned long long ga = (unsigned long long)(uintptr_t)gsrc;
    u32x4 g0;
    g0.x = 1u;                                                   // count = 1
    g0.y = (unsigned)__builtin_amdgcn_readfirstlane((int)ldsByteOff);
    g0.z = (unsigned)__builtin_amdgcn_readfirstlane((int)(unsigned)ga);
    g0.w = (unsigned)__builtin_amdgcn_readfirstlane(
               (int)(((unsigned)(ga >> 32) & 0x01FFFFFFu) | (2u << 30))); // type=2
    i32x8 g1;
    // data_size=2 (4B) | pad_enable | pad_interval=5 (64 DW) | pad_amount=3 (4 DW)
    g1[0] = (int)((2u << 16) | (1u << 20) | (5u << 22) | (3u << 25));
    g1[1] = (int)((384u & 0xFFFFu) << 16);        // tensor_dim0 = D_ (lo16)
    g1[2] = (int)((384u >> 16) | (1024u << 16));  // tensor_dim0 hi | tensor_dim1 lo
    g1[3] = (int)((1024u >> 16) | (64u << 16));   // tensor_dim1 hi | tile_dim0 = 64
    g1[4] = (int)8u;                              // tile_dim1 = 8 rows, tile_dim2 = 0
    g1[5] = (int)384u;                            // tensor_dim0_stride = D_
    g1[6] = 0;
    g1[7] = 0;
    i32x4 gz = {0, 0, 0, 0};
#if __clang_major__ >= 23
    i32x8 gz8 = {0, 0, 0, 0, 0, 0, 0, 0};
    __builtin_amdgcn_tensor_load_to_lds(g0, g1, gz, gz, gz8, 0);
#else
    __builtin_amdgcn_tensor_load_to_lds(g0, g1, gz, gz, 0);
#endif
}
#endif  // USE_TDM

// Fused: gram tile via V_WMMA_F32_16X16X4_F32 with TDM double-buffered LDS
// staging, then dist2 * exp(-kernel) reduction.
// Block = 256 threads = 8 waves; block tile = 64x64 of the (ti,tj) plane.
// Wave w: A row-group (w&3), B col-groups {2*(w>>2), 2*(w>>2)+1}.
__global__ __launch_bounds__(256) void PatchLoss_main_kernel(
    const float* __restrict__ z, const float* __restrict__ gt,
    const float* __restrict__ sigma, const float* __restrict__ sq,
    float* __restrict__ out) {
    __shared__ float ldsA[2][64][68];   // double buffer; 68 stride staggers banks
    __shared__ float ldsB[2][64][68];
    __shared__ float wave_part[8];

    const int b    = blockIdx.z;
    const int ti0  = blockIdx.x * 64;
    const int tj0  = blockIdx.y * 64;
    const int tid  = threadIdx.x;
    const int lane = tid & 31;
    const int wave = tid >> 5;
    const int wr     = wave & 3;   // A row group within tile
    const int wcpair = wave >> 2;  // 0/1 -> B col groups {0,1} or {2,3}
    const int half   = lane >> 4;  // hi half of wave
    const int mn     = lane & 15;

    const float* zb    = z + (size_t)b * T_ * D_;
    const float* zArow = zb + (size_t)ti0 * D_;
    const float* zBrow = zb + (size_t)tj0 * D_;

    // Cooperatively prefetch this block's 64KB gt_dT tile while gram runs.
    {
        const int pr = tid >> 2;
        const int pc = (tid & 3) * 16;
        const float* p = gt + ((size_t)(b * T_ + ti0 + pr) * T_ + tj0 + pc) * K_;
        __builtin_prefetch(p, 0, 0);  // -> global_prefetch_b8
    }

    v8f c0 = {};
    v8f c1 = {};

    const int aBase  = wr * 16 + mn;
    const int b0Base = wcpair * 32 + mn;
    const int b1Base = b0Base + 16;

#if USE_TDM
    // Preload K-chunk 0 into buffer 0: each wave DMA's its own 8-row slice.
    tdm_load_rows8(zArow + (size_t)(wave * 8) * D_,
                   (unsigned)(uintptr_t)&ldsA[0][wave * 8][0]);
    tdm_load_rows8(zBrow + (size_t)(wave * 8) * D_,
                   (unsigned)(uintptr_t)&ldsB[0][wave * 8][0]);
#endif

    for (int c = 0; c < (D_ / 64); ++c) {
        const int buf = c & 1;
#if USE_TDM
        if (c + 1 < (D_ / 64)) {
            const int kbn = (c + 1) * 64;
            tdm_load_rows8(zArow + (size_t)(wave * 8) * D_ + kbn,
                           (unsigned)(uintptr_t)&ldsA[buf ^ 1][wave * 8][0]);
            tdm_load_rows8(zBrow + (size_t)(wave * 8) * D_ + kbn,
                           (unsigned)(uintptr_t)&ldsB[buf ^ 1][wave * 8][0]);
            __builtin_amdgcn_s_wait_tensorcnt(2);  // chunk c landed; c+1 in flight
        } else {
            __builtin_amdgcn_s_wait_tensorcnt(0);  // final chunk landed
        }
        __syncthreads();
#else
        const int kb = c * 64;
        __syncthreads();
        for (int e = tid; e < 64 * 16; e += 256) {
            const int row = e >> 4;
            const int c4  = (e & 15) << 2;
            const float4 va = *(const float4*)(zArow + (size_t)row * D_ + kb + c4);
            const float4 vb = *(const float4*)(zBrow + (size_t)row * D_ + kb + c4);
            ldsA[buf][row][c4 + 0] = va.x; ldsA[buf][row][c4 + 1] = va.y;
            ldsA[buf][row][c4 + 2] = va.z; ldsA[buf][row][c4 + 3] = va.w;
            ldsB[buf][row][c4 + 0] = vb.x; ldsB[buf][row][c4 + 1] = vb.y;
            ldsB[buf][row][c4 + 2] = vb.z; ldsB[buf][row][c4 + 3] = vb.w;
        }
        __syncthreads();
#endif

#pragma unroll
        for (int k = 0; k < 64; k += 4) {
            // fp32 16x16x4 fragments: lanes 0-15 hold K = k,k+1;
            // lanes 16-31 hold K = k+2,k+3 (VGPR0/VGPR1).
            const int kk = k + half * 2;
            v2f af, bf0, bf1;
            af.x  = ldsA[buf][aBase][kk];  af.y  = ldsA[buf][aBase][kk + 1];
            bf0.x = ldsB[buf][b0Base][kk]; bf0.y = ldsB[buf][b0Base][kk + 1];
            bf1.x = ldsB[buf][b1Base][kk]; bf1.y = ldsB[buf][b1Base][kk + 1];
            c0 = __builtin_amdgcn_wmma_f32_16x16x4_f32(
                false, af, false, bf0, (short)0, c0, false, false);
            c1 = __builtin_amdgcn_wmma_f32_16x16x4_f32(
                false, af, false, bf1, (short)0, c1, false, false);
        }
#if USE_TDM
        __syncthreads();  // buffer buf free before it is DMA-overwritten
#endif
    }

    // Epilogue: dist2 = sq_i + sq_j - 2*gram; w = exp(-sum_k g^2/(2 sigma_k^2)).
    const float4 sg = *(const float4*)sigma;
    const float inv0 = 0.5f / (sg.x * sg.x);
    const float inv1 = 0.5f / (sg.y * sg.y);
    const float inv2 = 0.5f / (sg.z * sg.z);
    const float inv3 = 0.5f / (sg.w * sg.w);

    const float* sqb = sq + b * T_;
    const int jG0 = tj0 + wcpair * 32 + mn;
    const int jG1 = jG0 + 16;
    const float sqj0 = sqb[jG0];
    const float sqj1 = sqb[jG1];

    float partial = 0.0f;
#pragma unroll
    for (int v = 0; v < 8; ++v) {
        // C/D layout: lanes 0-15 -> M = v, lanes 16-31 -> M = v + 8.
        const int iG  = ti0 + wr * 16 + half * 8 + v;
        const float sqi = sqb[iG];
        const float d0 = sqi + sqj0 - 2.0f * c0[v];
        const float d1 = sqi + sqj1 - 2.0f * c1[v];
        const float4 g0 = *(const float4*)(gt + ((size_t)(b * T_ + iG) * T_ + jG0) * K_);
        const float4 g1 = *(const float4*)(gt + ((size_t)(b * T_ + iG) * T_ + jG1) * K_);
        const float ker0 = g0.x * g0.x * inv0 + g0.y * g0.y * inv1 +
                           g0.z * g0.z * inv2 + g0.w * g0.w * inv3;
        const float ker1 = g1.x * g1.x * inv0 + g1.y * g1.y * inv1 +
                           g1.z * g1.z * inv2 + g1.w * g1.w * inv3;
        partial += __expf(-ker0) * d0 + __expf(-ker1) * d1;
    }

#pragma unroll
    for (int off = 16; off > 0; off >>= 1) partial += __shfl_xor(partial, off, 32);
    if (lane == 0) wave_part[wave] = partial;
    __syncthreads();
    if (tid == 0) {
        float s = 0.0f;
#pragma unroll
        for (int w = 0; w < 8; ++w) s += wave_part[w];
        atomicAdd(out, s * (1.0f / ((float)B_ * (float)T_ * (float)T_)));
    }
}

extern "C" void kernel_launch(void* const* d_in, const int* in_sizes, int n_in,
                              void* d_out, int out_size, void* d_ws, size_t ws_size,
                              hipStream_t stream) {
    const float* z     = (const float*)d_in[0];   // [8,1024,384] f32
    const float* gt    = (const float*)d_in[1];   // [8,1024,1024,4] f32
    const float* sigma = (const float*)d_in[2];   // [4] f32
    float* out = (float*)d_out;                   // scalar
    float* sq  = (float*)d_ws;                    // [8*1024] row norms

    PatchLoss_zero_kernel<<<1, 1, 0, stream>>>(out);
    PatchLoss_rowsq_kernel<<<(B_ * T_) / 8, 256, 0, stream>>>(z, sq);
    dim3 grid(T_ / 64, T_ / 64, B_);
    PatchLoss_main_kernel<<<grid, 256, 0, stream>>>(z, gt, sigma, sq, out);
}